// SQCNN3D_47390669144598
// MI455X (gfx1250) — compile-verified
//
#include <hip/hip_runtime.h>
#include <hip/hip_bf16.h>

// ---------------- problem constants (from the reference) ----------------
#define NQ    6
#define D64   64          // 2^NQ
#define NFq   4
#define Gg    14
#define NSIDE 12
#define Pp    1728        // NSIDE^3
#define Bb    16
#define NCc   40
#define KTOT  41472       // NF * P * NQ
#define NSIM  (Pp*NFq*Bb) // 110592
#define IPT   (Pp*Bb)     // 27648 (p,b) pairs for the gram loss
#define IPBLK 108         // ceil(IPT/256)

// ---- split-K GEMM geometry ----
#define NTILE  3          // ceil(NC/16)
#define NCHUNK 162        // K split factor: 162 * 256 = 41472
#define KCHUNK 256
#define BK     64         // K per LDS stage
#define NSTAGE (KCHUNK/BK)

struct c32 { float x, y; };

typedef __attribute__((ext_vector_type(2))) float v2f;
typedef __attribute__((ext_vector_type(8))) float v8f;
typedef int v4i __attribute__((vector_size(16)));   // matches builtin proto

#define GAS __attribute__((address_space(1)))
#define LAS __attribute__((address_space(3)))

#if defined(__has_builtin)
#if __has_builtin(__builtin_amdgcn_global_load_async_to_lds_b128)
#define HAVE_ASYNC_LDS 1
#endif
#endif

__device__ __forceinline__ void stage16B(const float* g, float* l) {
#if defined(HAVE_ASYNC_LDS)
    // gfx1250 async DMA: copies 16 bytes from the global address in g to the LDS address in l, tracked by ASYNCcnt.
    // AS1 pointer via integer round-trip (bit-identical to generic for global);
    // AS3 pointer via explicit addrspacecast (LDS offsets are not bit-identical).
    __builtin_amdgcn_global_load_async_to_lds_b128(
        (GAS v4i*)(uintptr_t)g, (LAS v4i*)l, 0, 0);
#else
    *(float4*)l = *(const float4*)g;            // global_load_b128 + ds_store_b128
#endif
}

__device__ __forceinline__ void stage_wait() {
#if defined(HAVE_ASYNC_LDS)
#if defined(__has_builtin)
#if __has_builtin(__builtin_amdgcn_s_wait_asynccnt)
    __builtin_amdgcn_s_wait_asynccnt(0);
#else
    asm volatile("s_wait_asynccnt 0x0" ::: "memory");
#endif
#endif
#endif
    asm volatile("" ::: "memory");
}

// ---------------- kernel 1: per-thread 6-qubit statevector sim ----------------
// One thread == one (p,f,b) simulation; 64-complex state fully register-resident
// (all gate loops unrolled -> constant indices -> SROA to VGPRs).
__global__ __launch_bounds__(64)
void sqcnn_sim_kernel(const float* __restrict__ x,      // [B,14,14,14]
                      const float* __restrict__ theta,  // [NF,1,6,3]
                      float* __restrict__ probs,        // [P,NF,B,64]
                      float* __restrict__ feats)        // [B, NF*P*NQ]
{
    const int tid = blockIdx.x * blockDim.x + threadIdx.x;
    if (tid >= NSIM) return;
    const int b = tid % Bb;
    const int f = (tid / Bb) % NFq;
    const int p = tid / (Bb * NFq);
    const int pz = p % NSIDE;
    const int py = (p / NSIDE) % NSIDE;
    const int px = p / (NSIDE * NSIDE);
    const float* xb = x + (size_t)b * (Gg*Gg*Gg) + px*(Gg*Gg) + py*Gg + pz;

    c32 amp[D64];
#pragma unroll
    for (int d = 0; d < D64; ++d) amp[d] = {0.f, 0.f};
    amp[0].x = 1.f;

    // ---- 9 single-qubit Rot gates, angles from the patch (raster order) ----
#pragma unroll
    for (int k = 0; k < 9; ++k) {
        const int e0 = 3*k, e1 = 3*k + 1, e2 = 3*k + 2;
        const float phi = xb[(e0/9)*(Gg*Gg) + ((e0/3)%3)*Gg + (e0%3)];
        const float th  = xb[(e1/9)*(Gg*Gg) + ((e1/3)%3)*Gg + (e1%3)];
        const float om  = xb[(e2/9)*(Gg*Gg) + ((e2/3)%3)*Gg + (e2%3)];
        const float cc = cosf(0.5f*th), ss = sinf(0.5f*th);
        const float ap = -0.5f*(phi+om), am = 0.5f*(phi-om);
        const c32 ep = {cosf(ap), sinf(ap)};
        const c32 em = {cosf(am), sinf(am)};
        const c32 u00 = { ep.x*cc,  ep.y*cc};
        const c32 u01 = {-em.x*ss, -em.y*ss};
        const c32 u10 = { em.x*ss, -em.y*ss};   // conj(em)*s
        const c32 u11 = { ep.x*cc, -ep.y*cc};   // conj(ep)*c
        const int w  = k % NQ;
        const int st = 32 >> w;                 // qubit w is bit (5-w)
#pragma unroll
        for (int d = 0; d < D64; ++d) {
            if ((d & st) == 0) {
                const c32 a0 = amp[d], a1 = amp[d + st];
                amp[d]      = { u00.x*a0.x - u00.y*a0.y + u01.x*a1.x - u01.y*a1.y,
                                u00.x*a0.y + u00.y*a0.x + u01.x*a1.y + u01.y*a1.x };
                amp[d + st] = { u10.x*a0.x - u10.y*a0.y + u11.x*a1.x - u11.y*a1.y,
                                u10.x*a0.y + u10.y*a0.x + u11.x*a1.y + u11.y*a1.x };
            }
        }
    }

    // ---- 6 CROT gates (control i, target (i+1)%6), angles from theta[f] ----
    const float* tf = theta + (size_t)f * (NQ*3);   // L == 1
#pragma unroll
    for (int i = 0; i < NQ; ++i) {
        const float phi = tf[i*3+0], th = tf[i*3+1], om = tf[i*3+2];
        const float cc = cosf(0.5f*th), ss = sinf(0.5f*th);
        const float ap = -0.5f*(phi+om), am = 0.5f*(phi-om);
        const c32 ep = {cosf(ap), sinf(ap)};
        const c32 em = {cosf(am), sinf(am)};
        const c32 u00 = { ep.x*cc,  ep.y*cc};
        const c32 u01 = {-em.x*ss, -em.y*ss};
        const c32 u10 = { em.x*ss, -em.y*ss};
        const c32 u11 = { ep.x*cc, -ep.y*cc};
        const int cm = 32 >> i;
        const int tq = (i + 1) % NQ;
        const int tm = 32 >> tq;
#pragma unroll
        for (int d = 0; d < D64; ++d) {
            if ((d & cm) != 0 && (d & tm) == 0) {   // control set, target low half
                const c32 a0 = amp[d], a1 = amp[d + tm];
                amp[d]      = { u00.x*a0.x - u00.y*a0.y + u01.x*a1.x - u01.y*a1.y,
                                u00.x*a0.y + u00.y*a0.x + u01.x*a1.y + u01.y*a1.x };
                amp[d + tm] = { u10.x*a0.x - u10.y*a0.y + u11.x*a1.x - u11.y*a1.y,
                                u10.x*a0.y + u10.y*a0.x + u11.x*a1.y + u11.y*a1.x };
            }
        }
    }

    // ---- probs + Z expvals ----
    float pr[D64];
#pragma unroll
    for (int d = 0; d < D64; ++d) pr[d] = amp[d].x*amp[d].x + amp[d].y*amp[d].y;

    float* prow = probs + (size_t)tid * D64;        // tid order == [p][f][b]
#pragma unroll
    for (int d = 0; d < D64; ++d) prow[d] = pr[d];

    float* frow = feats + (size_t)b * KTOT + ((size_t)f * Pp + p) * NQ;
#pragma unroll
    for (int q = 0; q < NQ; ++q) {
        float e = 0.f;
#pragma unroll
        for (int d = 0; d < D64; ++d)
            e += (((d >> (5 - q)) & 1) ? -pr[d] : pr[d]);
        frow[q] = e;
    }
}

// -------- kernel 2: split-K logits GEMM, LDS-staged, V_WMMA_F32_16X16X4_F32 ----
// Grid (NTILE, NCHUNK), one wave per block. Each block: 16x16 partial tile over
// KCHUNK, staged through LDS in 16xBK tiles via async-to-LDS (coalesced rows),
// fragments read from LDS, accumulated with v_wmma_f32_16x16x4_f32.
__global__ __launch_bounds__(32)
void sqcnn_gemm_kernel(const float* __restrict__ feats,  // [16, KTOT]
                       const float* __restrict__ W,      // [40, KTOT]
                       float* __restrict__ part)         // [NCHUNK,NTILE,16,16]
{
    __shared__ __attribute__((aligned(16))) float ldsA[16*BK];
    __shared__ __attribute__((aligned(16))) float ldsB[16*BK];

    const int tile  = blockIdx.x;       // 0..2
    const int chunk = blockIdx.y;       // 0..NCHUNK-1
    const int lane  = threadIdx.x;      // 0..31
    const int h     = lane >> 4;        // K half: lanes 16-31 hold K=2,3
    const int r     = lane & 15;        // A row / B column within tile
    const int n     = tile * 16 + r;
    const float valid = (n < NCc) ? 1.0f : 0.0f;    // zero-pad cols 40..47
    const int  kbase  = chunk * KCHUNK;

    v8f acc = {};
    for (int s = 0; s < NSTAGE; ++s) {
        const int k0 = kbase + s * BK;
        // ---- stage A (feats rows 0..15) and B (W rows tile*16..+15) tiles ----
        // element e = i*32+lane: row = e>>4 (two rows per instr), 16B per lane,
        // 16 lanes cover 256B of one contiguous row -> fully coalesced.
#pragma unroll
        for (int i = 0; i < 8; ++i) {
            const int e   = i * 32 + lane;
            const int row = e >> 4;
            const int fo  = (e & 15) * 4;             // float offset in row
            const int nb  = tile * 16 + row;
            const int nbc = (nb < NCc) ? nb : (NCc - 1);
            stage16B(feats + (size_t)row * KTOT + k0 + fo, ldsA + row * BK + fo);
            stage16B(W     + (size_t)nbc * KTOT + k0 + fo, ldsB + row * BK + fo);
        }
        stage_wait();
        __syncthreads();

        // ---- consume: 16 WMMAs over this 16xBK stage ----
#pragma unroll
        for (int kk = 0; kk < BK; kk += 4) {
            const float* pa = ldsA + r * BK + kk + 2*h;   // A: M=r, K=2h,2h+1
            const float* pb = ldsB + r * BK + kk + 2*h;   // B: N=r, K=2h,2h+1
            v2f a, bv;
            a[0]  = pa[0];          a[1]  = pa[1];
            bv[0] = pb[0] * valid;  bv[1] = pb[1] * valid;
            acc = __builtin_amdgcn_wmma_f32_16x16x4_f32(
                false, a, false, bv, (short)0, acc, false, false);
        }
#if defined(HAVE_ASYNC_LDS)
        asm volatile("s_wait_dscnt 0x0" ::: "memory");  // LDS reads done before re-stage
#endif
        __syncthreads();
    }

    // ---- store 16x16 partial (C/D layout: vgpr j -> rows j and j+8) ----
    float* po = part + (((size_t)chunk * NTILE + tile) * 16) * 16;
#pragma unroll
    for (int j = 0; j < 8; ++j) {
        const int m = j + 8*h;
        po[m * 16 + r] = acc[j];
    }
}

// -------- kernel 3: deterministic split-K reduction + bias -> logits ----------
__global__ __launch_bounds__(256)
void sqcnn_logits_reduce(const float* __restrict__ part,
                         const float* __restrict__ bias,
                         float* __restrict__ out)
{
    const int t = blockIdx.x * 256 + threadIdx.x;
    if (t >= Bb * NCc) return;
    const int m = t / NCc, n = t % NCc;
    const int tile = n >> 4, col = n & 15;
    float s = 0.f;
    for (int c = 0; c < NCHUNK; ++c)                    // fixed order: deterministic
        s += part[(((size_t)c * NTILE + tile) * 16 + m) * 16 + col];
    out[m * NCc + n] = s + bias[n];
}

// -------- kernel 4: per-(p,b) 4x4 Gram, masked sum of squares ----------------
__global__ __launch_bounds__(256)
void sqcnn_ip_kernel(const float* __restrict__ probs, float* __restrict__ partials)
{
    __shared__ float sdata[256];
    const int t = blockIdx.x * 256 + threadIdx.x;
    float acc = 0.f;
    if (t < IPT) {
        const int b = t % Bb, p = t / Bb;
        const float* p0 = probs + (((size_t)p*NFq + 0)*Bb + b) * D64;
        const float* p1 = probs + (((size_t)p*NFq + 1)*Bb + b) * D64;
        const float* p2 = probs + (((size_t)p*NFq + 2)*Bb + b) * D64;
        const float* p3 = probs + (((size_t)p*NFq + 3)*Bb + b) * D64;
        float d01=0,d02=0,d03=0,d12=0,d13=0,d23=0;
#pragma unroll 8
        for (int d = 0; d < D64; ++d) {
            const float a = p0[d], bq = p1[d], c = p2[d], e = p3[d];
            d01 += a*bq; d02 += a*c; d03 += a*e;
            d12 += bq*c; d13 += bq*e; d23 += c*e;
        }
        acc = 2.f*(d01*d01 + d02*d02 + d03*d03 + d12*d12 + d13*d13 + d23*d23);
    }
    sdata[threadIdx.x] = acc;
    __syncthreads();
#pragma unroll
    for (int s = 128; s > 0; s >>= 1) {
        if (threadIdx.x < s) sdata[threadIdx.x] += sdata[threadIdx.x + s];
        __syncthreads();
    }
    if (threadIdx.x == 0) partials[blockIdx.x] = sdata[0];
}

// -------- kernel 5: deterministic final reduction -> loss --------------------
__global__ __launch_bounds__(128)
void sqcnn_reduce_kernel(const float* __restrict__ partials, float* __restrict__ out)
{
    __shared__ float sdata[128];
    sdata[threadIdx.x] = (threadIdx.x < IPBLK) ? partials[threadIdx.x] : 0.f;
    __syncthreads();
#pragma unroll
    for (int s = 64; s > 0; s >>= 1) {
        if (threadIdx.x < s) sdata[threadIdx.x] += sdata[threadIdx.x + s];
        __syncthreads();
    }
    // loss = sum * LAM / (B * NF*(NF-1)) = sum * 0.1 / 192
    if (threadIdx.x == 0) out[Bb * NCc] = sdata[0] * (0.1f / 192.0f);
}

// ---------------- host glue ----------------
extern "C" void kernel_launch(void* const* d_in, const int* in_sizes, int n_in,
                              void* d_out, int out_size, void* d_ws, size_t ws_size,
                              hipStream_t stream)
{
    const float* x     = (const float*)d_in[0];   // [16,14,14,14]
    const float* theta = (const float*)d_in[1];   // [4,1,6,3]
    const float* W     = (const float*)d_in[2];   // [40,41472]
    const float* bias  = (const float*)d_in[3];   // [40]
    float* out = (float*)d_out;                   // 640 logits + 1 loss
    float* ws  = (float*)d_ws;

    float* probs    = ws;                                      // NSIM*64
    float* feats    = probs    + (size_t)NSIM * D64;           // B*KTOT
    float* gpart    = feats    + (size_t)Bb * KTOT;            // NCHUNK*NTILE*256
    float* lpart    = gpart    + (size_t)NCHUNK * NTILE * 256; // IPBLK

    sqcnn_sim_kernel<<<NSIM / 64, 64, 0, stream>>>(x, theta, probs, feats);
    sqcnn_gemm_kernel<<<dim3(NTILE, NCHUNK), 32, 0, stream>>>(feats, W, gpart);
    sqcnn_logits_reduce<<<(Bb*NCc + 255)/256, 256, 0, stream>>>(gpart, bias, out);
    sqcnn_ip_kernel<<<IPBLK, 256, 0, stream>>>(probs, lpart);
    sqcnn_reduce_kernel<<<1, 128, 0, stream>>>(lpart, out);
}